// TVDAdvector_10660108829455
// MI455X (gfx1250) — compile-verified
//
#include <hip/hip_runtime.h>

typedef float v4f __attribute__((ext_vector_type(4)));
typedef int   v4i __attribute__((ext_vector_type(4)));

// ---------------------------------------------------------------------------
// Pass 1: per-link TVD (van Leer) flux.  4 links per thread via b128 loads.
// Streamed arrays use non-temporal loads (read exactly once); the random
// tracer gathers use regular loads so tracer (16MB) stays hot in the 192MB L2.
// Flux is stored with regular (RT) temporal hint because pass 2 re-reads it.
// ---------------------------------------------------------------------------
__global__ __launch_bounds__(256) void tvd_flux_kernel(
    const float* __restrict__ velocity,
    const float* __restrict__ tracer,
    const float* __restrict__ upwind,
    const int*  __restrict__ head,
    const int*  __restrict__ tail,
    float* __restrict__ flux,
    int n_links)
{
    const int i4   = blockIdx.x * blockDim.x + threadIdx.x;
    const int base = i4 * 4;
    if (base >= n_links) return;   // n_links % 4 == 0, so full vec4 always

    const v4f vel = __builtin_nontemporal_load((const v4f*)(velocity + base));
    const v4f upw = __builtin_nontemporal_load((const v4f*)(upwind   + base));
    const v4i hd  = __builtin_nontemporal_load((const v4i*)(head     + base));
    const v4i tl  = __builtin_nontemporal_load((const v4i*)(tail     + base));

    v4f fx;
#pragma unroll
    for (int j = 0; j < 4; ++j) {
        const float v      = vel[j];
        const float t_head = tracer[hd[j]];   // random gather, L2-resident
        const float t_tail = tracer[tl[j]];   // random gather, L2-resident
        const bool  pos    = (v >= 0.0f);
        const float center   = pos ? t_tail : t_head;
        const float downwind = pos ? t_head : t_tail;
        const float diff     = downwind - center;
        const float r  = (diff != 0.0f) ? (center - upw[j]) / diff : 0.0f;
        const float ar = fabsf(r);
        const float vl = (r + ar) / (1.0f + ar);       // van Leer limiter
        const float face = fmaf(0.5f * vl, diff, center);
        fx[j] = v * face;
    }
    *(v4f*)(flux + base) = fx;   // RT store: consumed by pass 2 from L2
}

// ---------------------------------------------------------------------------
// Pass 2: per-node divergence + tracer update.  links_at_node row is one
// aligned int4 -> single global_load_b128.  Flux gathers hit L2 (32MB buf).
// ---------------------------------------------------------------------------
__global__ __launch_bounds__(256) void tvd_update_kernel(
    const float* __restrict__ tracer,
    const float* __restrict__ cell_area,
    const int*  __restrict__ links_at_node,
    const float* __restrict__ flux,
    const float* __restrict__ dt_ptr,
    float* __restrict__ out,
    int n_nodes)
{
    const int i = blockIdx.x * blockDim.x + threadIdx.x;
    if (i >= n_nodes) return;

    const float dt = dt_ptr[0];   // uniform scalar load

    const v4i lk = __builtin_nontemporal_load((const v4i*)links_at_node + i);

    // Random gathers into L2-resident flux buffer (regular temporal hint).
    const float f0 = flux[lk.x];
    const float f1 = flux[lk.y];
    const float f2 = flux[lk.z];
    const float f3 = flux[lk.w];
    const float s  = ((f0 + f1) + f2) + f3;

    const float area = __builtin_nontemporal_load(cell_area + i);
    const float tr   = tracer[i];
    const float div  = (area != 0.0f) ? (s / area) : 0.0f;

    __builtin_nontemporal_store(fmaf(dt, div, tr), out + i);
}

extern "C" void kernel_launch(void* const* d_in, const int* in_sizes, int n_in,
                              void* d_out, int out_size, void* d_ws, size_t ws_size,
                              hipStream_t stream)
{
    const float* velocity      = (const float*)d_in[0];
    const float* tracer        = (const float*)d_in[1];
    const float* upwind        = (const float*)d_in[2];
    const float* cell_area     = (const float*)d_in[3];
    const int*   head          = (const int*)  d_in[4];
    const int*   tail          = (const int*)  d_in[5];
    const int*   links_at_node = (const int*)  d_in[6];
    const float* dt_ptr        = (const float*)d_in[7];

    const int n_links = in_sizes[0];   // 8,000,000
    const int n_nodes = in_sizes[1];   // 4,000,000

    float* flux = (float*)d_ws;        // n_links * 4 bytes = 32 MB scratch
    float* out  = (float*)d_out;

    const int block = 256;
    const int grid1 = (n_links / 4 + block - 1) / block;
    tvd_flux_kernel<<<grid1, block, 0, stream>>>(
        velocity, tracer, upwind, head, tail, flux, n_links);

    const int grid2 = (n_nodes + block - 1) / block;
    tvd_update_kernel<<<grid2, block, 0, stream>>>(
        tracer, cell_area, links_at_node, flux, dt_ptr, out, n_nodes);
}